// GCN_30468497998102
// MI455X (gfx1250) — compile-verified
//
#include <hip/hip_runtime.h>
#include <hip/hip_bf16.h>

// CDNA5 (gfx1250) GCN forward: 3x (WMMA-f32 GEMM -> normalized scatter-agg) with BN+ReLU.
// wave32; V_WMMA_F32_16X16X4_F32 for the dense x@W parts (fp32 exactness vs reference).

typedef __attribute__((ext_vector_type(2))) float v2f;
typedef __attribute__((ext_vector_type(8))) float v8f;

#define EPSV 1e-5f

// ---------------- degree / dinv ----------------
__global__ void k_init_deg(float* __restrict__ deg, int n) {
    int i = blockIdx.x * blockDim.x + threadIdx.x;
    if (i < n) deg[i] = 1.0f;  // self loop
}

__global__ void k_deg_scatter(const long long* __restrict__ dst, float* __restrict__ deg, int E) {
    int e = blockIdx.x * blockDim.x + threadIdx.x;
    if (e < E) atomicAdd(&deg[(int)dst[e]], 1.0f);
}

__global__ void k_dinv(float* __restrict__ deg, int n) {
    int i = blockIdx.x * blockDim.x + threadIdx.x;
    if (i < n) {
        float d = deg[i];
        deg[i] = (d > 0.0f) ? rsqrtf(d) : 0.0f;
    }
}

// zero-pad W[K,Nf] -> Wp[K,NPAD]
__global__ void k_pad_w(const float* __restrict__ W, float* __restrict__ Wp,
                        int K, int Nf, int Npad) {
    int idx = blockIdx.x * blockDim.x + threadIdx.x;
    if (idx >= K * Npad) return;
    int k = idx / Npad;
    int n = idx - k * Npad;
    Wp[idx] = (n < Nf) ? W[k * Nf + n] : 0.0f;
}

// ---------------- fp32 WMMA GEMM: O[M,Nf] = A[M,K] @ Wp[K,NPAD] ----------------
// K, NPAD compile-time; NPAD multiple of 16, B slab zero-padded so loads are unguarded.
// One wave per (nTile, chunk of mPerWave row tiles). B slab (16 cols x K) held in
// registers and reused across the row tiles. M must be a multiple of 16 (50000 = 3125*16).
// Fragment layouts per CDNA5 ISA 7.12.2:
//   A 16x4 f32 : lanes 0-15 hold M=lane, VGPR{0,1}=K{0,1}; lanes 16-31 hold K{2,3}.
//   B 4x16 f32 : lanes 0-15 hold N=lane, VGPR{0,1}=K{0,1}; lanes 16-31 hold K{2,3}.
//   C/D 16x16  : VGPR j -> row m0+j (lanes 0-15) / m0+8+j (lanes 16-31), col = lane%16.
template <int K, int NPAD>
__global__ void k_gemm_wmma_f32(const float* __restrict__ A, const float* __restrict__ Wp,
                                float* __restrict__ O, int M, int Nf, int mPerWave) {
    constexpr int kSteps = K / 4;
    const int wavesPerBlock = blockDim.x >> 5;
    const int wid = blockIdx.x * wavesPerBlock + (threadIdx.x >> 5);
    const int nTiles = NPAD >> 4;
    const int mTiles = M >> 4;
    const int mChunks = (mTiles + mPerWave - 1) / mPerWave;
    if (wid >= mChunks * nTiles) return;  // wave-uniform exit: EXEC all-1 for WMMA

    const int nT = wid % nTiles;
    const int mChunk = wid / nTiles;
    const int lane = threadIdx.x & 31;
    const int half = lane >> 4;      // 0: K rows {0,1} mod 4   1: K rows {2,3} mod 4
    const int l    = lane & 15;
    const int n0 = nT << 4;
    const int col = n0 + l;

    // ---- preload B slab into registers (compile-time offsets) ----
    v2f bf[kSteps];
    {
        const float* bp = Wp + (size_t)(2 * half) * NPAD + col;
#pragma unroll
        for (int kk = 0; kk < kSteps; ++kk) {
            bf[kk].x = bp[4 * kk * NPAD];
            bf[kk].y = bp[4 * kk * NPAD + NPAD];
        }
    }

    const bool cok = col < Nf;  // ragged Nf (C=40) handled only at the store

    for (int t = 0; t < mPerWave; ++t) {
        const int mT = mChunk * mPerWave + t;
        if (mT >= mTiles) break;             // uniform across the wave
        const int m0 = mT << 4;
        const float* ap = A + (size_t)(m0 + l) * K + 2 * half;  // 8B aligned (K even)

        v8f acc = {};
#pragma unroll
        for (int kk = 0; kk < kSteps; ++kk) {
            v2f a = *(const v2f*)(ap + 4 * kk);  // global_load_b64, immediate offset
            acc = __builtin_amdgcn_wmma_f32_16x16x4_f32(
                false, a, false, bf[kk], (short)0, acc, false, false);
        }

        if (cok) {
            const int r0 = m0 + half * 8;
#pragma unroll
            for (int j = 0; j < 8; ++j)
                O[(size_t)(r0 + j) * Nf + col] = acc[j];
        }
    }
}

// ---------------- aggregation ----------------
// agg[i,f] = h[i,f] * dinv[i]^2 (+ bias[f] if given)  -- self-loop term + accumulator init
__global__ void k_agg_init(const float* __restrict__ h, const float* __restrict__ dinv,
                           const float* __restrict__ bias, float* __restrict__ agg,
                           int n, int F) {
    int idx = blockIdx.x * blockDim.x + threadIdx.x;
    if (idx >= n * F) return;
    int i = idx / F;
    int f = idx - i * F;
    float di = dinv[i];
    float v = h[idx] * di * di;
    if (bias) v += bias[f];
    agg[idx] = v;
}

// one thread per (edge, float4 chunk): agg[dst] += h[src] * dinv[src]*dinv[dst]
__global__ void k_edge_scatter(const long long* __restrict__ src, const long long* __restrict__ dst,
                               const float* __restrict__ h, const float* __restrict__ dinv,
                               float* __restrict__ agg, int E, int F) {
    const int cpe = F >> 2;  // float4 chunks per edge (F = 64 or 40, both %4 == 0)
    int idx = blockIdx.x * blockDim.x + threadIdx.x;
    int total = E * cpe;
    if (idx >= total) return;
    int e = idx / cpe;
    int c = (idx - e * cpe) << 2;
    int s = (int)src[e];
    int d = (int)dst[e];
    float w = dinv[s] * dinv[d];
    const float4 hv = *(const float4*)(h + (size_t)s * F + c);  // 16B aligned: F%4==0, c%4==0
    float* out = agg + (size_t)d * F + c;
    atomicAdd(out + 0, hv.x * w);
    atomicAdd(out + 1, hv.y * w);
    atomicAdd(out + 2, hv.z * w);
    atomicAdd(out + 3, hv.w * w);
}

// o = relu( (agg + b - m) * rsqrt(v+eps) * g + be )
__global__ void k_bn_relu(const float* __restrict__ agg, const float* __restrict__ b,
                          const float* __restrict__ g, const float* __restrict__ be,
                          const float* __restrict__ m, const float* __restrict__ v,
                          float* __restrict__ o, int n, int F) {
    int idx = blockIdx.x * blockDim.x + threadIdx.x;
    if (idx >= n * F) return;
    int f = idx % F;
    float x = agg[idx] + b[f];
    x = (x - m[f]) * rsqrtf(v[f] + EPSV) * g[f] + be[f];
    o[idx] = fmaxf(x, 0.0f);
}

// ---------------- host orchestration ----------------
extern "C" void kernel_launch(void* const* d_in, const int* in_sizes, int n_in,
                              void* d_out, int out_size, void* d_ws, size_t ws_size,
                              hipStream_t stream) {
    const float*     x   = (const float*)d_in[0];
    const long long* ei  = (const long long*)d_in[1];   // int64, shape (2, E) flat
    const float*     W1  = (const float*)d_in[2];
    const float*     b1  = (const float*)d_in[3];
    const float*     g1  = (const float*)d_in[4];
    const float*     be1 = (const float*)d_in[5];
    const float*     m1  = (const float*)d_in[6];
    const float*     v1  = (const float*)d_in[7];
    const float*     W2  = (const float*)d_in[8];
    const float*     b2  = (const float*)d_in[9];
    const float*     g2  = (const float*)d_in[10];
    const float*     be2 = (const float*)d_in[11];
    const float*     m2  = (const float*)d_in[12];
    const float*     v2  = (const float*)d_in[13];
    const float*     W3  = (const float*)d_in[14];
    const float*     b3  = (const float*)d_in[15];

    const int Fin = 128, H = 64, C = 40, CP = 48;  // CP: W3 columns padded to 3 tiles
    const int N = in_sizes[0] / Fin;       // 50000 (multiple of 16)
    const int E = in_sizes[1] / 2;         // 1.6M
    const long long* srcI = ei;
    const long long* dstI = ei + E;

    float* dinv = (float*)d_ws;            // N floats (deg -> dinv in place)
    float* bufA = dinv + N;                // N*H floats
    float* bufB = bufA + (size_t)N * H;    // N*H floats
    float* W3p  = bufB + (size_t)N * H;    // H*CP floats (zero-padded W3)
    float* outp = (float*)d_out;           // N*C floats

    const int TB = 256;
    const int wavesPerBlock = TB / 32;
    const int mPerWave = 4;
    auto cdiv = [](long long a, long long b) { return (int)((a + b - 1) / b); };
    const int mChunks = cdiv(N / 16, mPerWave);

    // --- degree / dinv ---
    k_init_deg   <<<cdiv(N, TB), TB, 0, stream>>>(dinv, N);
    k_deg_scatter<<<cdiv(E, TB), TB, 0, stream>>>(dstI, dinv, E);
    k_dinv       <<<cdiv(N, TB), TB, 0, stream>>>(dinv, N);
    // --- pad W3 (64x40 -> 64x48) ---
    k_pad_w      <<<cdiv(H * CP, TB), TB, 0, stream>>>(W3, W3p, H, C, CP);

    // --- layer 1: h1 = x @ W1 ; agg ; BN+ReLU ---
    {
        int waves = mChunks * (H / 16);
        k_gemm_wmma_f32<128, 64><<<cdiv(waves, wavesPerBlock), TB, 0, stream>>>(x, W1, bufA, N, H, mPerWave);
        k_agg_init    <<<cdiv((long long)N * H, TB), TB, 0, stream>>>(bufA, dinv, nullptr, bufB, N, H);
        k_edge_scatter<<<cdiv((long long)E * (H / 4), TB), TB, 0, stream>>>(srcI, dstI, bufA, dinv, bufB, E, H);
        k_bn_relu     <<<cdiv((long long)N * H, TB), TB, 0, stream>>>(bufB, b1, g1, be1, m1, v1, bufA, N, H);
    }

    // --- layer 2: h2 = x2 @ W2 ; agg ; BN+ReLU ---
    {
        int waves = mChunks * (H / 16);
        k_gemm_wmma_f32<64, 64><<<cdiv(waves, wavesPerBlock), TB, 0, stream>>>(bufA, W2, bufB, N, H, mPerWave);
        k_agg_init    <<<cdiv((long long)N * H, TB), TB, 0, stream>>>(bufB, dinv, nullptr, bufA, N, H);
        k_edge_scatter<<<cdiv((long long)E * (H / 4), TB), TB, 0, stream>>>(srcI, dstI, bufB, dinv, bufA, E, H);
        k_bn_relu     <<<cdiv((long long)N * H, TB), TB, 0, stream>>>(bufA, b2, g2, be2, m2, v2, bufB, N, H);
    }

    // --- layer 3: h3 = x3 @ W3 (padded) ; agg (+bias) straight into d_out ---
    {
        int waves = mChunks * (CP / 16);
        k_gemm_wmma_f32<64, 48><<<cdiv(waves, wavesPerBlock), TB, 0, stream>>>(bufB, W3p, bufA, N, C, mPerWave);
        k_agg_init    <<<cdiv((long long)N * C, TB), TB, 0, stream>>>(bufA, dinv, b3, outp, N, C);
        k_edge_scatter<<<cdiv((long long)E * (C / 4), TB), TB, 0, stream>>>(srcI, dstI, bufA, dinv, outp, E, C);
    }

    (void)n_in; (void)out_size; (void)ws_size;
}